// TSALoss_85641647882503
// MI455X (gfx1250) — compile-verified
//
#include <hip/hip_runtime.h>

typedef __attribute__((ext_vector_type(2))) float v2f;
typedef __attribute__((ext_vector_type(8))) float v8f;

#define NB    8192
#define DIM   64
#define KNN   25
#define TOPK  25
#define FINF  3.4e38f

// ---------------- recon = sum((o-t)^2) ----------------
__global__ __launch_bounds__(256) void recon_kernel(const float* __restrict__ o,
                                                    const float* __restrict__ t,
                                                    float* __restrict__ acc, int n) {
  __shared__ float red[256];
  float s = 0.f;
  for (int i = blockIdx.x * blockDim.x + threadIdx.x; i < n; i += gridDim.x * blockDim.x) {
    float d = o[i] - t[i];
    s += d * d;
  }
  red[threadIdx.x] = s;
  __syncthreads();
  for (int off = 128; off > 0; off >>= 1) {
    if (threadIdx.x < off) red[threadIdx.x] += red[threadIdx.x + off];
    __syncthreads();
  }
  if (threadIdx.x == 0) atomicAdd(acc, red[0]);
}

// ---------------- sq[i] = ||raw_i||^2 ----------------
__global__ __launch_bounds__(256) void sqnorm_kernel(const float* __restrict__ raw,
                                                     float* __restrict__ sq) {
  int r = blockIdx.x * blockDim.x + threadIdx.x;
  const float4* p = (const float4*)(raw + r * DIM);
  float s = 0.f;
#pragma unroll
  for (int i = 0; i < DIM / 4; ++i) {
    float4 v = p[i];
    s += v.x * v.x + v.y * v.y + v.z * v.z + v.w * v.w;
  }
  sq[r] = s;
}

// ---------------- fused pairwise-d2 (WMMA f32) + per-row top-25 ----------------
// 512 threads = 16 waves. Block owns 16 query rows; wave w maintains top-K of row w.
__global__ __launch_bounds__(512) void dist_topk_kernel(const float* __restrict__ raw,
                                                        const float* __restrict__ sq,
                                                        int* __restrict__ nbrs) {
  __shared__ float Arows[16][DIM + 2];   // +2 pad: conflict-free fragment loads
  __shared__ float sqA[16];
  __shared__ float tileBuf[16][256];     // 16 rows x 256 cols of d2 per sweep

  const int tid  = threadIdx.x;
  const int lane = tid & 31;
  const int wave = tid >> 5;
  const int hi   = lane >> 4;            // 0/1 half-wave
  const int lo   = lane & 15;
  const int rowbase = blockIdx.x * 16;

  for (int e = tid; e < 16 * DIM; e += 512) {
    int r = e >> 6, c = e & 63;
    Arows[r][c] = raw[(rowbase + r) * DIM + c];
  }
  if (tid < 16) sqA[tid] = sq[rowbase + tid];
  __syncthreads();

  // A fragments: element (M=lo, K=4k+2*hi+{0,1}) — invariant across column sweep
  v2f af[16];
#pragma unroll
  for (int k = 0; k < 16; ++k) {
    af[k].x = Arows[lo][4 * k + 2 * hi];
    af[k].y = Arows[lo][4 * k + 2 * hi + 1];
  }
  float sqArr[8];
#pragma unroll
  for (int r = 0; r < 8; ++r) sqArr[r] = sqA[r + 8 * hi];

  // per-lane sorted (ascending) top-25
  float bd[TOPK];
  int   bi[TOPK];
#pragma unroll
  for (int k = 0; k < TOPK; ++k) { bd[k] = FINF; bi[k] = -1; }

  const int myrow = rowbase + wave;

  for (int t = 0; t < NB / 256; ++t) {
    const int colbase = t * 256 + wave * 16;   // this wave's 16-column tile
    const float* bptr = raw + (colbase + lo) * DIM + 2 * hi;  // B[k][n] = raw[col+n][k]

    // Preload ALL B fragments first: lets the 16 global_load_b64 issue as a
    // clause with one wait, then 16 back-to-back WMMAs (acc chaining is hazard-free).
    v2f bf[16];
#pragma unroll
    for (int k = 0; k < 16; ++k) {
      bf[k].x = bptr[4 * k];
      bf[k].y = bptr[4 * k + 1];
    }
    float sqn = sq[colbase + lo];

    v8f acc = {};
#pragma unroll
    for (int k = 0; k < 16; ++k)
      acc = __builtin_amdgcn_wmma_f32_16x16x4_f32(false, af[k], false, bf[k],
                                                  (short)0, acc, false, false);

#pragma unroll
    for (int r = 0; r < 8; ++r)
      tileBuf[r + 8 * hi][wave * 16 + lo] = sqArr[r] + sqn - 2.f * acc[r];
    __syncthreads();

    // wave w scans row w; lane handles cols c = lane (mod 32)
#pragma unroll 1
    for (int c = lane; c < 256; c += 32) {
      int j = t * 256 + c;
      float v = tileBuf[wave][c];
      if (j != myrow && v < bd[TOPK - 1]) {
        bd[TOPK - 1] = v; bi[TOPK - 1] = j;
#pragma unroll
        for (int k = TOPK - 1; k > 0; --k) {      // static-index bubble: stays in VGPRs
          if (bd[k] < bd[k - 1]) {
            float tv = bd[k]; bd[k] = bd[k - 1]; bd[k - 1] = tv;
            int   ti = bi[k]; bi[k] = bi[k - 1]; bi[k - 1] = ti;
          }
        }
      }
    }
    __syncthreads();
  }

  // 32-way merge of per-lane sorted lists via butterfly arg-min; pop winner's head
  for (int s = 0; s < KNN; ++s) {
    float v = bd[0]; int j = bi[0]; int wl = lane;
#pragma unroll
    for (int off = 16; off > 0; off >>= 1) {
      float ov = __shfl_xor(v, off, 32);
      int   oj = __shfl_xor(j, off, 32);
      int   ol = __shfl_xor(wl, off, 32);
      if (ov < v) { v = ov; j = oj; wl = ol; }
    }
    if (lane == 0) nbrs[myrow * KNN + s] = j;
    if (lane == wl) {
#pragma unroll
      for (int k = 0; k < TOPK - 1; ++k) { bd[k] = bd[k + 1]; bi[k] = bi[k + 1]; }
      bd[TOPK - 1] = FINF;
    }
  }
}

// ---------------- per-sample: C = Zc^T Zc (WMMA), power iteration, 2-2(u.v)^2 ----
__global__ __launch_bounds__(128) void tsa_kernel(const float* __restrict__ latent,
                                                  const float* __restrict__ raw,
                                                  const int* __restrict__ nbrs,
                                                  float* __restrict__ tsa_acc) {
  __shared__ float Mb[28][DIM];          // K padded 25 -> 28 (zero rows)
  __shared__ float Cb[DIM][DIM + 1];     // +1 pad: conflict-free matvec rows
  __shared__ float meanb[DIM];
  __shared__ float ub[DIM];
  __shared__ float uz[DIM];
  __shared__ float red[DIM];
  __shared__ float rnorm;
  __shared__ int   nb[KNN];

  const int tid  = threadIdx.x;
  const int lane = tid & 31;
  const int wave = tid >> 5;
  const int hi   = lane >> 4;
  const int lo   = lane & 15;
  const int b    = blockIdx.x;

  if (tid < KNN) nb[tid] = nbrs[b * KNN + tid];
  __syncthreads();

  for (int phase = 0; phase < 2; ++phase) {
    const float* src = (phase == 0) ? latent : raw;

    for (int e = tid; e < 28 * DIM; e += 128) {
      int r = e >> 6, c = e & 63;
      Mb[r][c] = (r < KNN) ? src[nb[r] * DIM + c] : 0.f;
    }
    __syncthreads();
    if (tid < DIM) {
      float s = 0.f;
      for (int r = 0; r < KNN; ++r) s += Mb[r][tid];
      meanb[tid] = s * (1.f / KNN);
    }
    __syncthreads();
    for (int e = tid; e < KNN * DIM; e += 128) {
      int r = e >> 6, c = e & 63;
      Mb[r][c] -= meanb[c];
    }
    __syncthreads();

    // C(64x64) = M^T M : 16 tiles across 4 waves, K-loop 0..27 step 4 (wave-uniform)
    for (int tile = wave; tile < 16; tile += 4) {
      int tm = tile >> 2, tn = tile & 3;

      // Preload the 7 A/B fragment pairs (ds_load clause, single dscnt wait),
      // then run the WMMA chain back-to-back.
      v2f a[7], bb[7];
#pragma unroll
      for (int s = 0; s < 7; ++s) {
        int k0 = 4 * s;
        a[s].x  = Mb[k0 + 2 * hi][tm * 16 + lo];
        a[s].y  = Mb[k0 + 2 * hi + 1][tm * 16 + lo];
        bb[s].x = Mb[k0 + 2 * hi][tn * 16 + lo];
        bb[s].y = Mb[k0 + 2 * hi + 1][tn * 16 + lo];
      }
      v8f acc = {};
#pragma unroll
      for (int s = 0; s < 7; ++s)
        acc = __builtin_amdgcn_wmma_f32_16x16x4_f32(false, a[s], false, bb[s],
                                                    (short)0, acc, false, false);
#pragma unroll
      for (int r = 0; r < 8; ++r)
        Cb[tm * 16 + r + 8 * hi][tn * 16 + lo] = acc[r];
    }
    __syncthreads();

    // power iteration for top eigenvector (scale by 1/denom irrelevant)
    if (tid < DIM) ub[tid] = 1.f;
    __syncthreads();
    for (int it = 0; it < 40; ++it) {
      float y = 0.f;
      if (tid < DIM) {
#pragma unroll 8
        for (int k = 0; k < DIM; ++k) y += Cb[tid][k] * ub[k];
        red[tid] = y * y;
      }
      __syncthreads();
      if (tid == 0) {
        float s = 0.f;
        for (int i = 0; i < DIM; ++i) s += red[i];
        rnorm = rsqrtf(s + 1e-30f);
      }
      __syncthreads();
      if (tid < DIM) ub[tid] = y * rnorm;
      __syncthreads();
    }
    if (phase == 0) {
      if (tid < DIM) uz[tid] = ub[tid];
      __syncthreads();
    }
  }

  // ||uu^T - vv^T||_F^2 = 2 - 2 (u.v)^2  for unit u,v
  if (tid == 0) {
    float dt = 0.f;
    for (int c = 0; c < DIM; ++c) dt += uz[c] * ub[c];
    atomicAdd(tsa_acc, 2.f - 2.f * dt * dt);
  }
}

// ---------------- combine ----------------
__global__ void finalize_kernel(const float* __restrict__ ws, float* __restrict__ out) {
  if (threadIdx.x == 0 && blockIdx.x == 0)
    out[0] = ws[0] * (1.f / (float)(NB * DIM)) + 0.1f * (ws[1] * (1.f / (float)NB));
}

extern "C" void kernel_launch(void* const* d_in, const int* in_sizes, int n_in,
                              void* d_out, int out_size, void* d_ws, size_t ws_size,
                              hipStream_t stream) {
  const float* outputs = (const float*)d_in[0];
  const float* targets = (const float*)d_in[1];
  const float* latent  = (const float*)d_in[2];
  const float* raw     = (const float*)d_in[3];
  float* out = (float*)d_out;

  float* wsf       = (float*)d_ws;
  float* recon_sum = wsf;          // [1]
  float* tsa_sum   = wsf + 1;      // [1]
  float* sq        = wsf + 2;      // [NB]
  int*   nbrs      = (int*)(wsf + 2 + NB);  // [NB*KNN]

  hipMemsetAsync(d_ws, 0, 2 * sizeof(float), stream);
  recon_kernel<<<512, 256, 0, stream>>>(outputs, targets, recon_sum, NB * DIM);
  sqnorm_kernel<<<NB / 256, 256, 0, stream>>>(raw, sq);
  dist_topk_kernel<<<NB / 16, 512, 0, stream>>>(raw, sq, nbrs);
  tsa_kernel<<<NB, 128, 0, stream>>>(latent, raw, nbrs, tsa_sum);
  finalize_kernel<<<1, 64, 0, stream>>>(wsf, out);
}